// ConcatDecoder_22179211117029
// MI455X (gfx1250) — compile-verified
//
#include <hip/hip_runtime.h>

typedef __attribute__((ext_vector_type(2))) float v2f;
typedef __attribute__((ext_vector_type(4))) float v4f;
typedef __attribute__((ext_vector_type(8))) float v8f;

#define NN   1024
#define FF   128
#define KK   4
#define CH   (2 * FF + KK)        // 260 channels per (i,j)
#define EPSV 1e-6f

// ---------------------------------------------------------------------------
// Kernel 1: row norms of z (1024 rows x 128). One wave32 per row.
// ---------------------------------------------------------------------------
__global__ void norms_kernel(const float* __restrict__ z,
                             float* __restrict__ norms) {
    int wave = (int)((blockIdx.x * blockDim.x + threadIdx.x) >> 5);
    int lane = (int)(threadIdx.x & 31);
    const float* row = z + wave * FF;
    float s = 0.0f;
#pragma unroll
    for (int c = 0; c < FF; c += 32) {
        float v = row[c + lane];
        s = fmaf(v, v, s);
    }
#pragma unroll
    for (int off = 16; off > 0; off >>= 1)
        s += __shfl_xor(s, off, 32);
    if (lane == 0) norms[wave] = sqrtf(s);
}

// ---------------------------------------------------------------------------
// Kernel 2: cos = (z @ z^T) / max(n_i * n_j, eps) via V_WMMA_F32_16X16X4_F32.
// One wave per 16x16 output tile; K-loop over 128 in steps of 4.
//
// A (16x4 f32) per-lane layout (ISA 7.12.2): lanes 0-15 -> K=k,k+1 ;
// lanes 16-31 -> K=k+2,k+3 ; M = lane&15. B (4x16) mirrors with N = lane&15.
// C/D: VGPR r holds row (i0 + 8*(lane>>4) + r), col (j0 + (lane&15)).
// ---------------------------------------------------------------------------
__global__ void cos_gemm_kernel(const float* __restrict__ z,
                                const float* __restrict__ norms,
                                float* __restrict__ cosout) {
    int waveId = (int)((blockIdx.x * blockDim.x + threadIdx.x) >> 5);
    int lane   = (int)(threadIdx.x & 31);
    int ti = waveId >> 6;          // 64 tiles per dim
    int tj = waveId & 63;
    int i0 = ti << 4;
    int j0 = tj << 4;

    int m    = lane & 15;
    int koff = (lane >> 4) << 1;   // 0 for lanes 0-15, 2 for lanes 16-31

    const float* arow = z + (i0 + m) * FF + koff;  // A: rows of tile i
    const float* brow = z + (j0 + m) * FF + koff;  // B: rows of tile j (z^T)

    v8f c = {};
#pragma unroll 8
    for (int k = 0; k < FF; k += 4) {
        v2f a = *(const v2f*)(arow + k);
        v2f b = *(const v2f*)(brow + k);
        c = __builtin_amdgcn_wmma_f32_16x16x4_f32(
                /*neg_a=*/false, a, /*neg_b=*/false, b,
                /*c_mod=*/(short)0, c, /*reuse_a=*/false, /*reuse_b=*/false);
    }

    int col = j0 + m;
    float nj = norms[col];
    int rbase = i0 + ((lane >> 4) << 3);
#pragma unroll
    for (int r = 0; r < 8; ++r) {
        int row = rbase + r;
        float ni = norms[row];
        float denom = fmaxf(ni * nj, EPSV);
        cosout[(size_t)row * NN + col] = c[r] / denom;
    }
}

// ---------------------------------------------------------------------------
// Kernel 3: adj[i][j][:] = concat(z[i], z[j], raw_adj[:,i,j]).
// Flat over N*N*65 float4 elements -> consecutive threads store consecutive
// 16B chunks (coalesced b128). Non-temporal: 1.09 GB >> 192 MB L2.
// ---------------------------------------------------------------------------
__global__ void adj_kernel(const float* __restrict__ z,
                           const float* __restrict__ raw_adj,
                           float* __restrict__ adj) {
    long long idx  = (long long)blockIdx.x * blockDim.x + threadIdx.x;
    int       c4   = (int)(idx % 65);        // float4 slot within 260-ch row
    long long pair = idx / 65;
    int       i    = (int)(pair >> 10);
    int       j    = (int)(pair & (NN - 1));

    v4f val;
    if (c4 < 32) {
        val = *(const v4f*)(z + i * FF + (c4 << 2));
    } else if (c4 < 64) {
        val = *(const v4f*)(z + j * FF + ((c4 - 32) << 2));
    } else {
        long long off = (long long)i * NN + j;
        const long long plane = (long long)NN * NN;
        val.x = raw_adj[off];
        val.y = raw_adj[off + plane];
        val.z = raw_adj[off + 2 * plane];
        val.w = raw_adj[off + 3 * plane];
    }
    __builtin_nontemporal_store(val, (v4f*)adj + idx);
}

// ---------------------------------------------------------------------------
extern "C" void kernel_launch(void* const* d_in, const int* in_sizes, int n_in,
                              void* d_out, int out_size, void* d_ws, size_t ws_size,
                              hipStream_t stream) {
    const float* z       = (const float*)d_in[0];   // [1024,128]
    const float* raw_adj = (const float*)d_in[1];   // [4,1024,1024]
    float* out   = (float*)d_out;
    float* adj   = out;                                       // [N,N,260]
    float* coso  = out + (size_t)NN * NN * CH;                // [N*N,1]
    float* norms = (float*)d_ws;                              // 1024 floats

    // norms: 1024 waves = 32768 threads -> 128 blocks of 256
    norms_kernel<<<(NN * 32) / 256, 256, 0, stream>>>(z, norms);

    // gemm: 4096 tiles, 1 wave each, 8 waves/block -> 512 blocks
    cos_gemm_kernel<<<4096 / 8, 256, 0, stream>>>(z, norms, coso);

    // adj: N*N*65 float4s = 68,157,440 threads -> 266,240 blocks (exact)
    long long total = (long long)NN * NN * 65;
    adj_kernel<<<(unsigned)(total / 256), 256, 0, stream>>>(z, raw_adj, adj);
}